// Add_Gryo_LCM_32323923870227
// MI455X (gfx1250) — compile-verified
//
#include <hip/hip_runtime.h>

typedef __attribute__((ext_vector_type(2))) float v2f;
typedef __attribute__((ext_vector_type(8))) float v8f;

#define N 64
#define LSTR 68  // 68*4 = 272 B row stride: rows stay 16B-aligned (B128 LDS ops legal)
                 // and column accesses map to banks (4*row + col) % 64 -> conflict-free
                 // for the 16-lane WMMA fragment reads.

// Wave-synchronous LDS fence: DS ops from one wave complete in order; this
// guarantees prior DS stores are visible to subsequent DS loads of the same
// wave, and the "memory" clobber stops the compiler reordering across it.
__device__ __forceinline__ void wave_lds_fence() {
  asm volatile("s_wait_dscnt 0" ::: "memory");
}

// ---------------------------------------------------------------------------
// Scalar full-matrix Cholesky (used only by the run-once bias kernel).
// ---------------------------------------------------------------------------
__device__ __forceinline__ void cholesky64_lds(float A[N][LSTR], int tid) {
  for (int j = 0; j < N; ++j) {
    float d   = A[j][j];
    float inv = rsqrtf(d);
    __syncthreads();
    if (tid > j && tid < N) A[tid][j] *= inv;
    if (tid == j)           A[j][j]    = d * inv;
    __syncthreads();
#pragma unroll
    for (int q = 0; q < 16; ++q) {
      int idx = tid * 16 + q;
      int i = idx >> 6, k = idx & 63;
      if (i > j && k > j) A[i][k] -= A[i][j] * A[k][j];
    }
    __syncthreads();
  }
}

// ---------------------------------------------------------------------------
// Kernel 1 (runs once, 1 block): bias_L = cholesky( expm( sym(bias) ) )
// expm via scaling-and-squaring: E = (Taylor_12(S/128))^(2^7)
// ---------------------------------------------------------------------------
__global__ void __launch_bounds__(256)
bias_prep_kernel(const float* __restrict__ bias, float* __restrict__ biasL) {
  __shared__ __align__(16) float sA[N][LSTR];
  __shared__ __align__(16) float sR[N][LSTR];
  __shared__ __align__(16) float sT[N][LSTR];
  const int tid = threadIdx.x;

#pragma unroll
  for (int q = 0; q < 16; ++q) {
    int idx = tid + 256 * q;
    int i = idx >> 6, k = idx & 63;
    sA[i][k] = 0.5f * (bias[i * N + k] + bias[k * N + i]) * 0.0078125f; // 2^-7
    sR[i][k] = (i == k) ? 1.0f : 0.0f;
  }
  __syncthreads();

  // Horner: for d = 12..1:  R = I + (A @ R) / d
  for (int dterm = 12; dterm >= 1; --dterm) {
    float invd = 1.0f / (float)dterm;
#pragma unroll
    for (int q = 0; q < 16; ++q) {
      int idx = tid * 16 + q;
      int i = idx >> 6, k = idx & 63;
      float acc = 0.0f;
      for (int j = 0; j < N; ++j) acc += sA[i][j] * sR[j][k];
      sT[i][k] = acc * invd + ((i == k) ? 1.0f : 0.0f);
    }
    __syncthreads();
#pragma unroll
    for (int q = 0; q < 16; ++q) {
      int idx = tid * 16 + q;
      sR[idx >> 6][idx & 63] = sT[idx >> 6][idx & 63];
    }
    __syncthreads();
  }

  // Square 7 times: R = R @ R
  for (int s = 0; s < 7; ++s) {
#pragma unroll
    for (int q = 0; q < 16; ++q) {
      int idx = tid * 16 + q;
      int i = idx >> 6, k = idx & 63;
      float acc = 0.0f;
      for (int j = 0; j < N; ++j) acc += sR[i][j] * sR[j][k];
      sT[i][k] = acc;
    }
    __syncthreads();
#pragma unroll
    for (int q = 0; q < 16; ++q) {
      int idx = tid * 16 + q;
      sR[idx >> 6][idx & 63] = sT[idx >> 6][idx & 63];
    }
    __syncthreads();
  }

  cholesky64_lds(sR, tid);
  __syncthreads();

#pragma unroll
  for (int q = 0; q < 16; ++q) {
    int idx = tid + 256 * q;
    int i = idx >> 6, k = idx & 63;
    biasL[idx] = (i >= k) ? sR[i][k] : 0.0f;
  }
}

// ---------------------------------------------------------------------------
// Kernel 2 (one workgroup = 8 wave32 per matrix):
//   blocked Cholesky (16x16 blocks, rank-16 trailing updates on WMMA fp32)
//   -> combine with bias factor -> out = M @ M^T on WMMA fp32.
// ---------------------------------------------------------------------------
__global__ void __launch_bounds__(256)
batch_chol_gemm_kernel(const float* __restrict__ x,
                       const float* __restrict__ biasL,
                       float* __restrict__ out) {
  __shared__ __align__(16) float sA[N][LSTR];
  __shared__ float sInvD[16];      // reciprocals of current panel's L_pp diagonal
  const int tid  = threadIdx.x;
  const int lane = tid & 31;
  const int wave = tid >> 5;
  const int half = lane >> 4;
  const int l16  = lane & 15;
  const size_t b = blockIdx.x;
  const float* xb = x + b * (size_t)(N * N);
  float* ob = out + b * (size_t)(N * N);

  // ---- load x_b: global_load_b128 + ds_store_b128 (rows are 16B aligned) ----
#pragma unroll
  for (int q = 0; q < 4; ++q) {
    int vidx = tid + 256 * q;                 // float4 index
    float4 v = ((const float4*)xb)[vidx];
    int e = vidx * 4;
    *(float4*)&sA[e >> 6][e & 63] = v;
  }
  __syncthreads();

  // ---- blocked right-looking Cholesky, block size 16 ----
#pragma unroll
  for (int p = 0; p < 4; ++p) {
    const int pb = p * 16;

    // (1) factor the 16x16 diagonal block: wave 0 only, wave-synchronous
    //     (DS ops in-order within a wave + explicit s_wait_dscnt fences).
    //     Lane owns half-row: r = lane>>1, cols c0..c0+7.
    if (wave == 0) {
      const int r  = lane >> 1;
      const int c0 = (lane & 1) * 8;
      for (int j = 0; j < 16; ++j) {
        wave_lds_fence();
        float d   = sA[pb + j][pb + j];
        float inv = rsqrtf(d);               // == 1 / L[j][j]
        if ((lane & 1) == (j >> 3)) {        // this lane owns column j piece
          if (r > j)       sA[pb + r][pb + j] *= inv;
          else if (r == j) { sA[pb + j][pb + j] = d * inv; sInvD[j] = inv; }
        }
        wave_lds_fence();
        float lrj = sA[pb + r][pb + j];
#pragma unroll
        for (int s = 0; s < 8; ++s) {
          int c = c0 + s;
          if (r > j && c > j)
            sA[pb + r][pb + c] -= lrj * sA[pb + c][pb + j];
        }
      }
      wave_lds_fence();
    }
    __syncthreads();

    if (p < 3) {
      // (2) panel triangular solve: row gi solves X * L_pp^T = A[gi, pb:pb+16]
      const int nrows = 48 - pb;
      if (tid < nrows) {
        const int gi = pb + 16 + tid;
        float xr[16];
#pragma unroll
        for (int c = 0; c < 16; ++c) {
          float acc = sA[gi][pb + c];
          for (int d2 = 0; d2 < c; ++d2)
            acc -= xr[d2] * sA[pb + c][pb + d2];
          xr[c] = acc * sInvD[c];
          sA[gi][pb + c] = xr[c];
        }
      }
      __syncthreads();

      // (3) trailing update (lower-tri blocks only):
      //     A[bi,bk] -= L[bi,p] @ L[bk,p]^T, each tile = 4 x wmma_f32_16x16x4.
      //     f32 WMMA has no A/B negate (NEG = CNeg only), so accumulate the
      //     product into 0 and subtract in VALU.
      {
        int t = 0;
#pragma unroll
        for (int bi = p + 1; bi < 4; ++bi) {
#pragma unroll
          for (int bk = p + 1; bk <= bi; ++bk) {
            if (t == wave) {                 // uniform per wave: EXEC stays all-1s
              const float* arow = &sA[bi * 16 + l16][pb];
              const float* brow = &sA[bk * 16 + l16][pb];
              v8f prod = {};
#pragma unroll
              for (int kk = 0; kk < 4; ++kk) {
                const int kb = kk * 4 + half * 2;
                v2f a, bf;
                a.x  = arow[kb];  a.y  = arow[kb + 1];
                bf.x = brow[kb];  bf.y = brow[kb + 1];
                prod = __builtin_amdgcn_wmma_f32_16x16x4_f32(
                    false, a, false, bf, (short)0, prod, false, false);
              }
#pragma unroll
              for (int rr = 0; rr < 8; ++rr) {
                const int m = bi * 16 + rr + half * 8;
                const int n = bk * 16 + l16;
                sA[m][n] -= prod[rr];
              }
            }
            ++t;
          }
        }
      }
      __syncthreads();
    }
  }

  // ---- combine with bias factor: M = tril(L,-1)+tril(Lb,-1)+diag(dL*dLb) ----
  {
    const int ci = tid >> 2;               // row
    const int cc = (tid & 3) * 16;         // col base (16B aligned in LDS)
#pragma unroll
    for (int q4 = 0; q4 < 4; ++q4) {
      const int kbase = cc + q4 * 4;
      float4 lb = ((const float4*)biasL)[tid * 4 + q4];
      float4 l  = *(float4*)&sA[ci][kbase];
      float4 m;
      float* lp = (float*)&l; float* lbp = (float*)&lb; float* mp = (float*)&m;
#pragma unroll
      for (int cpt = 0; cpt < 4; ++cpt) {
        int k = kbase + cpt;
        mp[cpt] = (ci > k) ? (lp[cpt] + lbp[cpt])
                           : ((ci == k) ? (lp[cpt] * lbp[cpt]) : 0.0f);
      }
      *(float4*)&sA[ci][kbase] = m;
    }
  }
  __syncthreads();

  // ---- out = M @ M^T: 16 output tiles, 2 per wave, 16 WMMA each ----
#pragma unroll
  for (int t = 0; t < 2; ++t) {
    const int tile = wave * 2 + t;
    const int mi = tile >> 2, ni = tile & 3;
    const float* arow = &sA[mi * 16 + l16][0];
    const float* brow = &sA[ni * 16 + l16][0];
    v8f c = {};
#pragma unroll
    for (int kk = 0; kk < 16; ++kk) {
      const int kb = kk * 4 + half * 2;
      v2f a, bf;
      a.x  = arow[kb];  a.y  = arow[kb + 1];
      bf.x = brow[kb];  bf.y = brow[kb + 1];
      c = __builtin_amdgcn_wmma_f32_16x16x4_f32(false, a, false, bf,
                                                (short)0, c, false, false);
    }
#pragma unroll
    for (int r = 0; r < 8; ++r) {
      const int m = mi * 16 + r + half * 8;
      const int n = ni * 16 + l16;
      ob[m * N + n] = c[r];
    }
  }
}

// ---------------------------------------------------------------------------
extern "C" void kernel_launch(void* const* d_in, const int* in_sizes, int n_in,
                              void* d_out, int out_size, void* d_ws, size_t ws_size,
                              hipStream_t stream) {
  const float* x    = (const float*)d_in[0];  // [B,64,64] fp32
  const float* bias = (const float*)d_in[1];  // [64,64]  fp32
  float* out   = (float*)d_out;               // [B,64,64] fp32
  float* biasL = (float*)d_ws;                // 64*64 floats scratch

  const int Bn = in_sizes[0] / (N * N);

  bias_prep_kernel<<<1, 256, 0, stream>>>(bias, biasL);
  batch_chol_gemm_kernel<<<Bn, 256, 0, stream>>>(x, biasL, out);
}